// GraphConv_10969346474352
// MI455X (gfx1250) — compile-verified
//
#include <hip/hip_runtime.h>
#include <hip/hip_bf16.h>

typedef __attribute__((ext_vector_type(2))) float v2f;
typedef __attribute__((ext_vector_type(8))) float v8f;

#define D 128            // D_IN == D_OUT == 128
#define NROWS 100000
#define NEDGES 1600000

// ---------------------------------------------------------------------------
// Phase 1: support = X @ W   (fp32 WMMA, 16x16 tile per wave, K-loop of 4)
//   A (16x4 f32): lane l<16 -> M=l, holds K={k0,k0+1}; lane>=16 -> M=l-16, K={k0+2,k0+3}
//   B (4x16 f32): lane l<16 -> N=l, K={k0,k0+1};       lane>=16 -> N=l-16, K={k0+2,k0+3}
//   D (16x16 f32, 8 VGPRs): vgpr v, lane l -> M = v + 8*(l/16), N = l%16
// ---------------------------------------------------------------------------
__global__ void gc_gemm_wmma(const float* __restrict__ x,
                             const float* __restrict__ W,
                             float* __restrict__ support) {
    const int rowBlock = blockIdx.x;           // 16 output rows per block
    const int wave     = threadIdx.x >> 5;     // 8 waves -> 8 N-tiles (128 cols)
    const int lane     = threadIdx.x & 31;
    const int half     = lane >> 4;            // 0: K lo pair, 1: K hi pair
    const int l16      = lane & 15;

    const int row0 = rowBlock * 16;
    const int col0 = wave * 16;

    const float* aRow = x + (size_t)(row0 + l16) * D;   // this lane's A row
    const float* bCol = W + (size_t)(col0 + l16);       // this lane's B column

    v8f c = {};
    #pragma unroll 4
    for (int k0 = 0; k0 < D; k0 += 4) {
        const int ka = k0 + 2 * half;
        // A: contiguous float2 along K  -> global_load_b64
        v2f a = *(const v2f*)(aRow + ka);
        // B: two row-strided loads of W[k][n] (coalesced 64B per half-wave,
        // W is 64KB -> lives in WGP$/L2 after first touch)
        v2f b;
        b.x = bCol[(size_t)ka * D];
        b.y = bCol[(size_t)(ka + 1) * D];
        // (neg_a, A, neg_b, B, c_mod, C, reuse_a, reuse_b)
        c = __builtin_amdgcn_wmma_f32_16x16x4_f32(false, a, false, b,
                                                  (short)0, c, false, false);
    }

    float* outBase = support + (size_t)row0 * D + col0;
    #pragma unroll
    for (int v = 0; v < 8; ++v) {
        const int m = v + 8 * half;
        outBase[(size_t)m * D + l16] = c[v];
    }
}

// ---------------------------------------------------------------------------
// Phase 2a: out[n][d] = bias[d]   (float4 per thread, fully coalesced)
// ---------------------------------------------------------------------------
__global__ void gc_init_bias(const float* __restrict__ bias,
                             float* __restrict__ out) {
    const int i = blockIdx.x * blockDim.x + threadIdx.x;   // 0 .. N*32-1
    const int d4 = i & 31;                                 // 32 float4 per row
    float4 bv = ((const float4*)bias)[d4];
    ((float4*)out)[i] = bv;
}

// ---------------------------------------------------------------------------
// Phase 2b: out[row[e]] += vals[e] * support[col[e]]
//   One wave per edge: lane handles a float4 slice (512B coalesced gather),
//   4x global_atomic_add_f32 scatter. row/col/val are wave-uniform -> SMEM.
// ---------------------------------------------------------------------------
__global__ void gc_scatter(const float* __restrict__ support,
                           const int* __restrict__ rowIdx,
                           const int* __restrict__ colIdx,
                           const float* __restrict__ vals,
                           float* __restrict__ out) {
    const int e    = blockIdx.x * 8 + (threadIdx.x >> 5);
    const int lane = threadIdx.x & 31;

    const int   r = rowIdx[e];
    const int   c = colIdx[e];
    const float v = vals[e];

    float4 m = ((const float4*)(support + (size_t)c * D))[lane];
    float* dst = out + (size_t)r * D + lane * 4;
    atomicAdd(dst + 0, m.x * v);
    atomicAdd(dst + 1, m.y * v);
    atomicAdd(dst + 2, m.z * v);
    atomicAdd(dst + 3, m.w * v);
}

// ---------------------------------------------------------------------------
// inputs: [0]=x f32[N*128], [1]=W f32[128*128], [2]=b f32[128],
//         [3]=row i32[E], [4]=col i32[E], [5]=vals f32[E]
// out: f32[N*128]; ws: support f32[N*128] = 51.2 MB
// ---------------------------------------------------------------------------
extern "C" void kernel_launch(void* const* d_in, const int* in_sizes, int n_in,
                              void* d_out, int out_size, void* d_ws, size_t ws_size,
                              hipStream_t stream) {
    const float* x    = (const float*)d_in[0];
    const float* W    = (const float*)d_in[1];
    const float* bias = (const float*)d_in[2];
    const int*   row  = (const int*)d_in[3];
    const int*   col  = (const int*)d_in[4];
    const float* vals = (const float*)d_in[5];
    float* out     = (float*)d_out;
    float* support = (float*)d_ws;

    // 1) support = X @ W       (6250 row-blocks x 8 waves)
    gc_gemm_wmma<<<NROWS / 16, 256, 0, stream>>>(x, W, support);
    // 2) out = broadcast(bias) (N*32 float4 stores)
    gc_init_bias<<<(NROWS * 32) / 256, 256, 0, stream>>>(bias, out);
    // 3) out[row] += val * support[col]  (wave per edge, atomics)
    gc_scatter<<<NEDGES / 8, 256, 0, stream>>>(support, row, col, vals, out);
}